// GeneralizedIoU_10067403341970
// MI455X (gfx1250) — compile-verified
//
#include <hip/hip_runtime.h>
#include <math.h>

// ---------------------------------------------------------------------------
// GIoU + BCE fused loss, MI455X (gfx1250).
// Memory-bound streaming reduction: 160 MB in / 4 B out -> ~6.9 us @ 23.3 TB/s.
// CDNA5 path: double-buffered GLOBAL_LOAD_ASYNC_TO_LDS_B128 staging tracked
// with ASYNCcnt (s_wait_asynccnt), hardware transcendentals for the math.
// ---------------------------------------------------------------------------

#define THREADS          128                     // 4 waves (wave32)
#define ROWS_PER_THREAD  4
#define TILE_ROWS        (THREADS * ROWS_PER_THREAD)   // 512 rows / tile
#define TILE_FLOATS      (TILE_ROWS * 5)               // 2560 floats = 10240 B
#define VEC4_PER_THREAD  (TILE_FLOATS / 4 / THREADS)   // 5 x b128 per thread/array
#define MAX_BLOCKS       2048

__device__ __forceinline__ float fast_sigmoid(float x) {
    // 1 / (1 + exp(-x)) via v_exp_f32 (base-2) + IEEE divide
    float e = __builtin_amdgcn_exp2f(-1.4426950408889634f * x);
    return 1.0f / (1.0f + e);
}

// Per-row contribution: (1 - giou) + 0.2 * bce   (mean taken at the end)
__device__ __forceinline__ float row_contrib(const float* pr, const float* tg) {
    const float logit = pr[0];
    const float p0 = fast_sigmoid(pr[1]);
    const float p1 = fast_sigmoid(pr[2]);
    const float p2 = fast_sigmoid(pr[3]);
    const float p3 = fast_sigmoid(pr[4]);
    const float label = tg[0];
    const float t0 = tg[1], t1 = tg[2], t2 = tg[3], t3 = tg[4];

    const float pred_area   = fabsf(p2 - p0) * fabsf(p3 - p1);
    const float target_area = (t2 - t0) * (t3 - t1);

    const float x_p1 = fminf(p2, p0), x_p2 = fmaxf(p2, p0);
    const float y_p1 = fminf(p1, p3), y_p2 = fmaxf(p1, p3);

    const float x1 = fmaxf(x_p1, t0), x2 = fminf(x_p2, t2);
    const float y1 = fmaxf(y_p1, t1), y2 = fminf(y_p2, t3);
    float overlap = (x2 - x1) * (y2 - y1);
    overlap = (overlap < 0.0f) ? 0.0f : overlap;

    const float x1c = fminf(x_p1, t0), x2c = fmaxf(x_p2, t2);
    const float y1c = fminf(y_p1, t1), y2c = fmaxf(y_p2, t3);
    const float enclose = (x2c - x1c) * (y2c - y1c);

    const float uni  = target_area + pred_area - overlap;
    const float giou = overlap / uni - (enclose - uni) / enclose;

    // bce = max(l,0) - l*y + log1p(exp(-|l|)); |l| >= 0 so 1+e in (1,2]: stable
    const float a      = fabsf(logit);
    const float e      = __builtin_amdgcn_exp2f(-1.4426950408889634f * a);
    const float log1pe = 0.6931471805599453f * __builtin_amdgcn_logf(1.0f + e);
    const float bce    = fmaxf(logit, 0.0f) - logit * label + log1pe;

    return (1.0f - giou) + 0.2f * bce;
}

__global__ __launch_bounds__(THREADS)
void giou_bce_partial(const float* __restrict__ preds,
                      const float* __restrict__ target,
                      int N, double* __restrict__ blockSums) {
    // Double-buffered staging tiles + reduction scratch: 40 KB + 1 KB LDS.
    __shared__ float  ldsP[2][TILE_FLOATS];
    __shared__ float  ldsT[2][TILE_FLOATS];
    __shared__ double red[THREADS];

    const int tid      = threadIdx.x;
    const int numTiles = N / TILE_ROWS;

    float acc = 0.0f;

    // Issue one tile's worth of async b128 global->LDS loads (ASYNCcnt += 10/wave-lane set).
    auto issue = [&](int tile, int buf) {
        const float* gp = preds  + (size_t)tile * TILE_FLOATS;
        const float* gt = target + (size_t)tile * TILE_FLOATS;
#pragma unroll
        for (int k = 0; k < VEC4_PER_THREAD; ++k) {
            const int v4 = tid + k * THREADS;
            unsigned           lp = (unsigned)(size_t)(&ldsP[buf][v4 * 4]);
            unsigned           lt = (unsigned)(size_t)(&ldsT[buf][v4 * 4]);
            unsigned long long ap = (unsigned long long)(size_t)(gp + v4 * 4);
            unsigned long long at = (unsigned long long)(size_t)(gt + v4 * 4);
            asm volatile("global_load_async_to_lds_b128 %0, %1, off"
                         :: "v"(lp), "v"(ap) : "memory");
            asm volatile("global_load_async_to_lds_b128 %0, %1, off"
                         :: "v"(lt), "v"(at) : "memory");
        }
    };

    int tile = blockIdx.x;
    int buf  = 0;
    if (tile < numTiles) {
        issue(tile, 0);
        for (;;) {
            const int  next    = tile + gridDim.x;
            const bool hasNext = next < numTiles;
            if (hasNext) {
                issue(next, buf ^ 1);
                // 10 newest (next tile) may remain outstanding; current tile done.
                asm volatile("s_wait_asynccnt 0xa" ::: "memory");
            } else {
                asm volatile("s_wait_asynccnt 0x0" ::: "memory");
            }
            __syncthreads();   // current tile visible block-wide

            const float* lp = &ldsP[buf][tid * (ROWS_PER_THREAD * 5)];
            const float* lt = &ldsT[buf][tid * (ROWS_PER_THREAD * 5)];
#pragma unroll
            for (int j = 0; j < ROWS_PER_THREAD; ++j)
                acc += row_contrib(lp + 5 * j, lt + 5 * j);

            __syncthreads();   // all reads done before this buffer is rewritten
            if (!hasNext) break;
            tile = next;
            buf ^= 1;
        }
    }

    // Tail rows not covered by full tiles: scalar grid-stride path.
    for (int r = numTiles * TILE_ROWS + (int)(blockIdx.x * blockDim.x) + tid;
         r < N; r += (int)(gridDim.x * blockDim.x)) {
        float pr[5], tg[5];
#pragma unroll
        for (int k = 0; k < 5; ++k) {
            pr[k] = preds[(size_t)r * 5 + k];
            tg[k] = target[(size_t)r * 5 + k];
        }
        acc += row_contrib(pr, tg);
    }

    // Deterministic block tree-reduce in double.
    red[tid] = (double)acc;
    __syncthreads();
    for (int s = THREADS / 2; s > 0; s >>= 1) {
        if (tid < s) red[tid] += red[tid + s];
        __syncthreads();
    }
    if (tid == 0) blockSums[blockIdx.x] = red[0];
}

__global__ __launch_bounds__(256)
void final_reduce(const double* __restrict__ blockSums, int nBlocks,
                  float* __restrict__ out, int N) {
    __shared__ double red[256];
    double s = 0.0;
    for (int i = threadIdx.x; i < nBlocks; i += 256) s += blockSums[i];
    red[threadIdx.x] = s;
    __syncthreads();
    for (int st = 128; st > 0; st >>= 1) {
        if ((int)threadIdx.x < st) red[threadIdx.x] += red[threadIdx.x + st];
        __syncthreads();
    }
    if (threadIdx.x == 0) out[0] = (float)(red[0] / (double)N);
}

extern "C" void kernel_launch(void* const* d_in, const int* in_sizes, int n_in,
                              void* d_out, int out_size, void* d_ws, size_t ws_size,
                              hipStream_t stream) {
    const float* preds  = (const float*)d_in[0];
    const float* target = (const float*)d_in[1];
    const int N = in_sizes[0] / 5;

    double* blockSums = (double*)d_ws;

    const int numTiles = N / TILE_ROWS;
    int blocks = MAX_BLOCKS;
    if (numTiles > 0 && blocks > numTiles) blocks = numTiles;
    const int wsCap = (int)(ws_size / sizeof(double));
    if (wsCap > 0 && blocks > wsCap) blocks = wsCap;
    if (blocks < 1) blocks = 1;

    giou_bce_partial<<<blocks, THREADS, 0, stream>>>(preds, target, N, blockSums);
    final_reduce<<<1, 256, 0, stream>>>(blockSums, blocks, (float*)d_out, N);
}